// SelectiveSSM_85083302133980
// MI455X (gfx1250) — compile-verified
//
#include <hip/hip_runtime.h>
#include <hip/hip_bf16.h>
#include <stdint.h>

// Problem constants (from reference)
#define BATCH   4
#define SEQLEN  4096
#define DMODEL  1024
#define DSTATE  16

// Chunked-scan tiling
#define NCHUNK  32                    // time chunks
#define CHUNKL  (SEQLEN / NCHUNK)     // 128 timesteps per chunk
#define TILE_D  256                   // d-channels per block
#define WAVE_D  32                    // d-channels per wave (wave32)
#define NWAVE   (TILE_D / WAVE_D)     // 8 waves per block
#define SUB_T   32                    // timesteps per TDM sub-tile
#define NSUB    (CHUNKL / SUB_T)      // 4 sub-tiles per chunk

#if defined(__has_builtin)
#if __has_builtin(__builtin_amdgcn_tensor_load_to_lds)
#define USE_TDM 1
#endif
#endif
#ifndef USE_TDM
#define USE_TDM 0
#endif

typedef unsigned int v4u __attribute__((ext_vector_type(4)));
typedef int          v8i __attribute__((ext_vector_type(8)));
typedef int          v4i __attribute__((ext_vector_type(4)));

__device__ __forceinline__ float softplusf(float v) {
  return (v > 20.0f) ? v : log1pf(__expf(v));
}

// Issue one TDM 2-D tile load: SUB_T rows x WAVE_D floats, row stride DMODEL
// floats in global memory, packed contiguously into LDS at lds_dst.
__device__ __forceinline__ void stage_subtile(float* lds_dst, const float* gsrc,
                                              int lane) {
#if USE_TDM
  (void)lane;
  unsigned long long ga = (unsigned long long)(uintptr_t)gsrc;
  unsigned lds_off = (unsigned)(uintptr_t)lds_dst;  // low 32 bits = LDS offset
  v4u g0;
  g0[0] = 1u;  // count=1 (valid descriptor), user mode, no gather
  g0[1] = (unsigned)__builtin_amdgcn_readfirstlane((int)lds_off);
  g0[2] = (unsigned)__builtin_amdgcn_readfirstlane((int)(unsigned)ga);
  g0[3] = ((unsigned)__builtin_amdgcn_readfirstlane((int)(unsigned)(ga >> 32)) &
           0x1FFFFFFu) | (2u << 30);  // global_addr[56:32] | type=2 (image)
  v8i g1;
  g1[0] = (int)(2u << 16);                    // data_size=4B (code 2), no mask
  g1[1] = (int)((unsigned)WAVE_D << 16);      // tensor_dim0[15:0]  = 32
  g1[2] = (int)((unsigned)SUB_T  << 16);      // tensor_dim1[15:0]  = 32
  g1[3] = (int)((unsigned)WAVE_D << 16);      // tile_dim0          = 32
  g1[4] = (int)SUB_T;                         // tile_dim1=32, tile_dim2=0
  g1[5] = (int)DMODEL;                        // tensor_dim0_stride lo32 = 1024
  g1[6] = 0;
  g1[7] = 0;
  v4i z4 = {0, 0, 0, 0};
  asm volatile("" ::: "memory");
#if __clang_major__ >= 23
  v8i z8 = {0, 0, 0, 0, 0, 0, 0, 0};
  __builtin_amdgcn_tensor_load_to_lds(g0, g1, z4, z4, z8, 0);
#else
  __builtin_amdgcn_tensor_load_to_lds(g0, g1, z4, z4, 0);
#endif
  asm volatile("" ::: "memory");
#else
  // Fallback: coalesced lane copy (LDS ops from same wave are in-order).
  #pragma unroll
  for (int tt = 0; tt < SUB_T; ++tt)
    lds_dst[tt * WAVE_D + lane] = gsrc[(size_t)tt * DMODEL + lane];
#endif
}

__device__ __forceinline__ void stage_wait(bool more_pending) {
#if USE_TDM
  if (more_pending) __builtin_amdgcn_s_wait_tensorcnt(1);
  else              __builtin_amdgcn_s_wait_tensorcnt(0);
  asm volatile("" ::: "memory");
#else
  (void)more_pending;
#endif
}

// ---------------------------------------------------------------------------
// Phase 1: per-(b, d, chunk) local scan with zero initial state.
// Writes chunk-local end state s_loc[b][ch][n][d] to ws.
// ---------------------------------------------------------------------------
__global__ __launch_bounds__(256) void ssm_local_scan(
    const float* __restrict__ x, const float* __restrict__ Bm,
    const float* __restrict__ logd, float* __restrict__ sloc) {
  __shared__ float tile[NWAVE][2][SUB_T][WAVE_D];
  const int tx   = threadIdx.x;
  const int wid  = tx >> 5, lane = tx & 31;
  const int d    = blockIdx.x * TILE_D + tx;
  const int b    = blockIdx.z, ch = blockIdx.y;
  const int t0   = ch * CHUNKL;
  const int dw   = blockIdx.x * TILE_D + wid * WAVE_D;
  const float* xbase = x + ((size_t)b * SEQLEN + t0) * DMODEL + dw;

  const float delta = softplusf(logd[d]);
  float Adec[DSTATE], cin[DSTATE];
  #pragma unroll
  for (int n = 0; n < DSTATE; ++n) {
    float Bv = Bm[d * DSTATE + n];
    float a  = __expf(-delta * Bv);
    Adec[n]  = a;
    cin[n]   = (1.0f - a) * Bv;
  }
  float s[DSTATE];
  #pragma unroll
  for (int n = 0; n < DSTATE; ++n) s[n] = 0.0f;

  stage_subtile(&tile[wid][0][0][0], xbase, lane);
  stage_subtile(&tile[wid][1][0][0], xbase + (size_t)SUB_T * DMODEL, lane);
  #pragma unroll
  for (int k = 0; k < NSUB; ++k) {
    stage_wait(k + 1 < NSUB);
    const float* lt = &tile[wid][k & 1][0][0];
    #pragma unroll 4
    for (int tt = 0; tt < SUB_T; ++tt) {
      float xv = lt[tt * WAVE_D + lane];
      #pragma unroll
      for (int n = 0; n < DSTATE; ++n)
        s[n] = fmaf(s[n], Adec[n], cin[n] * xv);
    }
    if (k + 2 < NSUB)
      stage_subtile(&tile[wid][k & 1][0][0],
                    xbase + (size_t)(k + 2) * SUB_T * DMODEL, lane);
  }

  const size_t base = (size_t)(b * NCHUNK + ch) * DSTATE * DMODEL + d;
  #pragma unroll
  for (int n = 0; n < DSTATE; ++n)
    sloc[base + (size_t)n * DMODEL] = s[n];
}

// ---------------------------------------------------------------------------
// Phase 2: sequential combine across chunks (one thread per (b, d, n)).
// Rewrites ws in place: ws[b][ch][n][d] becomes the state at chunk START.
// Emits the final state into sfinal (tail of d_out).
// ---------------------------------------------------------------------------
__global__ __launch_bounds__(256) void ssm_chunk_combine(
    const float* __restrict__ state0, const float* __restrict__ Bm,
    const float* __restrict__ logd, float* __restrict__ ws,
    float* __restrict__ sfinal) {
  const int tid = blockIdx.x * 256 + threadIdx.x;
  const int d   = tid & (DMODEL - 1);
  const int n   = (tid >> 10) & (DSTATE - 1);
  const int b   = tid >> 14;

  const float delta = softplusf(logd[d]);
  const float Bv    = Bm[d * DSTATE + n];
  const float P     = __expf(-delta * Bv * (float)CHUNKL);  // A^CHUNKL

  float cur = state0[((size_t)b * DMODEL + d) * DSTATE + n];
  for (int ch = 0; ch < NCHUNK; ++ch) {
    const size_t idx = (size_t)(b * NCHUNK + ch) * DSTATE * DMODEL +
                       (size_t)n * DMODEL + d;
    float tmp = ws[idx];   // local end-state of chunk ch
    ws[idx]   = cur;       // state at START of chunk ch
    cur = fmaf(cur, P, tmp);
  }
  sfinal[((size_t)b * DMODEL + d) * DSTATE + n] = cur;
}

// ---------------------------------------------------------------------------
// Phase 3: final scan with correct chunk start-states; emits y.
// ---------------------------------------------------------------------------
__global__ __launch_bounds__(256) void ssm_final_scan(
    const float* __restrict__ x, const float* __restrict__ Bm,
    const float* __restrict__ Cm, const float* __restrict__ logd,
    const float* __restrict__ sstart, float* __restrict__ y) {
  __shared__ float tile[NWAVE][2][SUB_T][WAVE_D];
  const int tx   = threadIdx.x;
  const int wid  = tx >> 5, lane = tx & 31;
  const int d    = blockIdx.x * TILE_D + tx;
  const int b    = blockIdx.z, ch = blockIdx.y;
  const int t0   = ch * CHUNKL;
  const int dw   = blockIdx.x * TILE_D + wid * WAVE_D;
  const float* xbase = x + ((size_t)b * SEQLEN + t0) * DMODEL + dw;

  const float delta = softplusf(logd[d]);
  float Adec[DSTATE], cin[DSTATE], Cv[DSTATE];
  #pragma unroll
  for (int n = 0; n < DSTATE; ++n) {
    float Bv = Bm[d * DSTATE + n];
    float a  = __expf(-delta * Bv);
    Adec[n]  = a;
    cin[n]   = (1.0f - a) * Bv;
    Cv[n]    = Cm[d * DSTATE + n];
  }
  const size_t base = (size_t)(b * NCHUNK + ch) * DSTATE * DMODEL + d;
  float s[DSTATE];
  #pragma unroll
  for (int n = 0; n < DSTATE; ++n) s[n] = sstart[base + (size_t)n * DMODEL];

  stage_subtile(&tile[wid][0][0][0], xbase, lane);
  stage_subtile(&tile[wid][1][0][0], xbase + (size_t)SUB_T * DMODEL, lane);
  #pragma unroll
  for (int k = 0; k < NSUB; ++k) {
    stage_wait(k + 1 < NSUB);
    const float* lt = &tile[wid][k & 1][0][0];
    #pragma unroll 4
    for (int tt = 0; tt < SUB_T; ++tt) {
      float xv  = lt[tt * WAVE_D + lane];
      float acc = 0.0f;
      #pragma unroll
      for (int n = 0; n < DSTATE; ++n) {
        s[n] = fmaf(s[n], Adec[n], cin[n] * xv);
        acc  = fmaf(s[n], Cv[n], acc);
      }
      y[((size_t)b * SEQLEN + t0 + k * SUB_T + tt) * DMODEL + d] = acc;
    }
    if (k + 2 < NSUB)
      stage_subtile(&tile[wid][k & 1][0][0],
                    xbase + (size_t)(k + 2) * SUB_T * DMODEL, lane);
  }
}

extern "C" void kernel_launch(void* const* d_in, const int* in_sizes, int n_in,
                              void* d_out, int out_size, void* d_ws,
                              size_t ws_size, hipStream_t stream) {
  (void)in_sizes; (void)n_in; (void)out_size; (void)ws_size;
  const float* x    = (const float*)d_in[0];  // (b, s, d)
  const float* st0  = (const float*)d_in[1];  // (b, d, n)
  const float* Bm   = (const float*)d_in[2];  // (d, n)
  const float* Cm   = (const float*)d_in[3];  // (d, n)
  const float* logd = (const float*)d_in[4];  // (d,)
  float* out    = (float*)d_out;                               // y, then state
  float* sfinal = out + (size_t)BATCH * SEQLEN * DMODEL;       // (b, d, n)
  float* ws     = (float*)d_ws;  // NCHUNK chunk states: 8 MB

  dim3 grid(DMODEL / TILE_D, NCHUNK, BATCH);
  ssm_local_scan<<<grid, 256, 0, stream>>>(x, Bm, logd, ws);
  ssm_chunk_combine<<<(BATCH * DMODEL * DSTATE) / 256, 256, 0, stream>>>(
      st0, Bm, logd, ws, sfinal);
  ssm_final_scan<<<grid, 256, 0, stream>>>(x, Bm, Cm, logd, ws, out);
}